// DecoderV1_12240656794071
// MI455X (gfx1250) — compile-verified
//
#include <hip/hip_runtime.h>
#include <stdint.h>

#define NB 16               // batch
#define NPIX 409600         // 640*640
#define NCH 5
#define KTOP 100
#define BINS 4096
#define BPB 50              // streaming blocks per batch
#define CHUNK (NPIX / BPB)  // 8192 elements per block
#define STAGE_ELEMS 1024    // 256 threads * 4 floats per iteration
#define NITERS (CHUNK / STAGE_ELEMS)  // 8
#define NSTAGE 4            // async pipeline depth (per wave: 4 x 512B in flight)
#define CAP 1024
#define IOU_THR_C 0.5f

// ---------------- CDNA5 async global->LDS helpers ----------------
__device__ __forceinline__ void async_ld_b128(const float* g, void* lds) {
    // LDS byte address = low 32 bits of the generic (flat) shared pointer.
    unsigned l = (unsigned)(unsigned long long)lds;
    asm volatile("global_load_async_to_lds_b128 %0, %1, off"
                 :: "v"(l), "v"(g)
                 : "memory");
}
__device__ __forceinline__ void wait_async_n(int n) {
    switch (n) {
    case 0:  asm volatile("s_wait_asynccnt 0" ::: "memory"); break;
    case 1:  asm volatile("s_wait_asynccnt 1" ::: "memory"); break;
    case 2:  asm volatile("s_wait_asynccnt 2" ::: "memory"); break;
    default: asm volatile("s_wait_asynccnt 3" ::: "memory"); break;
    }
}

// Pass 1: per-batch 4096-bin histogram of top-12 float bits (positive scores only),
// streaming scores through a 4-deep async global->LDS pipeline (ASYNCcnt).
__global__ __launch_bounds__(256) void k_hist(const float* __restrict__ preds,
                                              unsigned* __restrict__ hist) {
    __shared__ unsigned lhist[BINS];
    __shared__ __align__(16) float stage[NSTAGE][STAGE_ELEMS];  // b128 needs 16B LDS alignment
    const int tid = threadIdx.x;
    const int b = blockIdx.x / BPB;
    const int blk = blockIdx.x % BPB;
    const float* scores = preds + (size_t)b * NCH * NPIX;   // channel 0
    const size_t base = (size_t)blk * CHUNK;

    for (int i = tid; i < BINS; i += 256) lhist[i] = 0u;
    __syncthreads();

    // prologue: stage iterations 0..NSTAGE-2
#pragma unroll
    for (int p = 0; p < NSTAGE - 1; ++p) {
        async_ld_b128(scores + base + (size_t)p * STAGE_ELEMS + (size_t)tid * 4,
                      &stage[p][tid * 4]);
    }
#pragma unroll
    for (int it = 0; it < NITERS; ++it) {
        const int buf = it & (NSTAGE - 1);
        if (it + NSTAGE - 1 < NITERS) {
            async_ld_b128(scores + base + (size_t)(it + NSTAGE - 1) * STAGE_ELEMS + (size_t)tid * 4,
                          &stage[(it + NSTAGE - 1) & (NSTAGE - 1)][tid * 4]);
        }
        // async loads complete in order -> iteration 'it' has landed in LDS
        const int rem = NITERS - 1 - it;
        wait_async_n(rem < (NSTAGE - 1) ? rem : (NSTAGE - 1));
        float4 v = *reinterpret_cast<const float4*>(&stage[buf][tid * 4]);
        const float f[4] = {v.x, v.y, v.z, v.w};
#pragma unroll
        for (int k = 0; k < 4; ++k) {
            const int bits = __float_as_int(f[k]);
            if (bits > 0) {  // positive float <=> score > 0
                atomicAdd(&lhist[((unsigned)bits) >> 19], 1u);
            }
        }
    }
    __syncthreads();
    unsigned* gh = hist + (size_t)b * BINS;
    for (int i = tid; i < BINS; i += 256) {
        const unsigned c = lhist[i];
        if (c) atomicAdd(&gh[i], c);
    }
}

__global__ void k_zero(unsigned* __restrict__ p, int n) {
    int i = blockIdx.x * blockDim.x + threadIdx.x;
    if (i < n) p[i] = 0u;
}

// Pass 2: find highest bin T such that count(bin >= T) >= KTOP (T=0 if fewer).
__global__ void k_thresh(const unsigned* __restrict__ hist,
                         unsigned* __restrict__ thrBin) {
    const int b = blockIdx.x;
    if (threadIdx.x == 0) {
        const unsigned* h = hist + (size_t)b * BINS;
        unsigned acc = 0;
        int t = 0;
        for (int i = BINS - 1; i >= 0; --i) {
            acc += h[i];
            if (acc >= KTOP) { t = i; break; }
        }
        thrBin[b] = (unsigned)t;
    }
}

// Pass 3: collect candidate (scoreBits, index) pairs at/above threshold bin.
// Rereads the score channel, which pass 1 left resident in the 192MB L2.
__global__ __launch_bounds__(256) void k_collect(const float* __restrict__ preds,
                                                 const unsigned* __restrict__ thrBin,
                                                 unsigned* __restrict__ counter,
                                                 uint2* __restrict__ cand) {
    const int tid = threadIdx.x;
    const int b = blockIdx.x / BPB;
    const int blk = blockIdx.x % BPB;
    const unsigned T = thrBin[b];
    const float* scores = preds + (size_t)b * NCH * NPIX;
    const size_t base = (size_t)blk * CHUNK;
#pragma unroll
    for (int it = 0; it < NITERS; ++it) {
        const unsigned e0 = (unsigned)(base + (size_t)it * STAGE_ELEMS + (size_t)tid * 4);
        float4 v = *reinterpret_cast<const float4*>(scores + e0);
        const float f[4] = {v.x, v.y, v.z, v.w};
#pragma unroll
        for (int k = 0; k < 4; ++k) {
            const int bits = __float_as_int(f[k]);
            if (bits > 0 && ((((unsigned)bits) >> 19) >= T)) {
                const unsigned p = atomicAdd(&counter[b], 1u);
                if (p < CAP) cand[(size_t)b * CAP + p] = make_uint2((unsigned)bits, e0 + (unsigned)k);
            }
        }
    }
}

// Pass 4: per batch — sort candidates (score desc, idx asc), take top 100,
// gather boxes, greedy NMS (IoU > 0.5), write [100,5] with zeroed suppressed rows.
__global__ __launch_bounds__(256) void k_nms(const float* __restrict__ preds,
                                             const unsigned* __restrict__ counter,
                                             const uint2* __restrict__ cand,
                                             float* __restrict__ out) {
    const int b = blockIdx.x;
    const int tid = threadIdx.x;
    __shared__ unsigned long long key[CAP];
    __shared__ float bx[KTOP][4];
    __shared__ float sc[KTOP];
    __shared__ int keep[KTOP];

    unsigned cnt = counter[b];
    if (cnt > CAP) cnt = CAP;
    for (int i = tid; i < CAP; i += 256) {
        if (i < (int)cnt) {
            const uint2 c = cand[(size_t)b * CAP + i];
            // composite: score bits (desc) then ~idx (desc == idx asc)
            key[i] = ((unsigned long long)c.x << 32) | (unsigned long long)(unsigned)(~c.y);
        } else {
            key[i] = 0ull;  // bits==0 => invalid slot
        }
    }
    __syncthreads();

    // bitonic sort, descending
    for (int k = 2; k <= CAP; k <<= 1) {
        for (int j = k >> 1; j > 0; j >>= 1) {
            for (int i = tid; i < CAP; i += 256) {
                const int ixj = i ^ j;
                if (ixj > i) {
                    const unsigned long long a = key[i];
                    const unsigned long long c = key[ixj];
                    const bool desc = ((i & k) == 0);
                    if (desc ? (a < c) : (a > c)) { key[i] = c; key[ixj] = a; }
                }
            }
            __syncthreads();
        }
    }

    if (tid < KTOP) {
        const unsigned long long kk = key[tid];
        const unsigned bits = (unsigned)(kk >> 32);
        const int valid = (bits != 0u);
        sc[tid] = __uint_as_float(bits);
        const unsigned idx = ~((unsigned)(kk & 0xFFFFFFFFull));
        const float* pb = preds + (size_t)b * NCH * NPIX;
#pragma unroll
        for (int c = 0; c < 4; ++c)
            bx[tid][c] = valid ? pb[(size_t)(1 + c) * NPIX + idx] : 0.0f;
        keep[tid] = valid;
    }
    __syncthreads();

    for (int i = 0; i < KTOP - 1; ++i) {
        if (tid > i && tid < KTOP && keep[i] && keep[tid]) {
            const float x1 = fmaxf(bx[i][0], bx[tid][0]);
            const float y1 = fmaxf(bx[i][1], bx[tid][1]);
            const float x2 = fminf(bx[i][2], bx[tid][2]);
            const float y2 = fminf(bx[i][3], bx[tid][3]);
            const float w = fmaxf(x2 - x1, 0.0f);
            const float h = fmaxf(y2 - y1, 0.0f);
            const float inter = w * h;
            const float ai = (bx[i][2] - bx[i][0]) * (bx[i][3] - bx[i][1]);
            const float aj = (bx[tid][2] - bx[tid][0]) * (bx[tid][3] - bx[tid][1]);
            const float uni = ai + aj - inter;
            const float iou = inter / uni;
            if (iou > IOU_THR_C) keep[tid] = 0;
        }
        __syncthreads();
    }

    if (tid < KTOP) {
        const int kp = keep[tid];
        float* o = out + (size_t)b * KTOP * 5 + (size_t)tid * 5;
        o[0] = kp ? sc[tid] : 0.0f;
#pragma unroll
        for (int c = 0; c < 4; ++c) o[1 + c] = kp ? bx[tid][c] : 0.0f;
    }
}

// ---------------- launcher ----------------
extern "C" void kernel_launch(void* const* d_in, const int* in_sizes, int n_in,
                              void* d_out, int out_size, void* d_ws, size_t ws_size,
                              hipStream_t stream) {
    (void)in_sizes; (void)n_in; (void)out_size; (void)ws_size;
    const float* preds = (const float*)d_in[0];
    float* out = (float*)d_out;

    // ws layout (uints): hist[NB*BINS] | thrBin[NB] | counter[NB] | cand[NB*CAP] (uint2)
    unsigned* ws = (unsigned*)d_ws;
    unsigned* hist = ws;
    unsigned* thrBin = ws + (size_t)NB * BINS;
    unsigned* counter = thrBin + NB;
    uint2* cand = (uint2*)(ws + (size_t)NB * BINS + 2 * NB);

    const int zn = NB * BINS + 2 * NB;
    k_zero<<<(zn + 255) / 256, 256, 0, stream>>>(ws, zn);
    k_hist<<<NB * BPB, 256, 0, stream>>>(preds, hist);
    k_thresh<<<NB, 64, 0, stream>>>(hist, thrBin);
    k_collect<<<NB * BPB, 256, 0, stream>>>(preds, thrBin, counter, cand);
    k_nms<<<NB, 256, 0, stream>>>(preds, counter, cand, out);
}